// BreakoutPlatformPosition_44461501448964
// MI455X (gfx1250) — compile-verified
//
#include <hip/hip_runtime.h>

#ifndef __has_builtin
#define __has_builtin(x) 0
#endif

// Wave32 ballot: predicate -> 32-bit mask in an SGPR (CDNA5 is wave32-only).
__device__ __forceinline__ unsigned int wave_ballot32(bool p) {
#if __has_builtin(__builtin_amdgcn_ballot_w32)
    return __builtin_amdgcn_ballot_w32(p);
#else
    return (unsigned int)__ballot(p);   // wave32: low 32 bits valid
#endif
}

// observations: (32, 16, 3, 210, 160) float32
// We need channel 0, row 189 of each (b, e): 512 rows x 160 pixels.
// Row stride in elements = 3*210*160 = 100800; row offset = 189*160 = 30240.
#define ROW_STRIDE 100800
#define ROW_OFFSET 30240
#define W 160

__global__ __launch_bounds__(256)
void BreakoutPlatformPosition_kernel(const float* __restrict__ obs,
                                     int* __restrict__ out,
                                     int n_rows) {
    const int lane = threadIdx.x & 31;
    const int wave = threadIdx.x >> 5;
    const int row  = blockIdx.x * 8 + wave;   // one wave32 per row
    if (row >= n_rows) return;                // whole-wave uniform exit

    // Thresholds: computed in double (as Python does), rounded to f32 (JAX weak typing).
    const float LO = (float)(100.0 / 255.0 - 0.15);
    const float HI = (float)(200.0 / 255.0 + 0.15);

    const float* __restrict__ p = obs + (size_t)row * ROW_STRIDE + ROW_OFFSET;

    // Build 160-bit mask: pixel index = chunk*32 + lane -> bit `lane` of word[chunk].
    // Each chunk is one fully-coalesced 128B wave load.
    unsigned int wmask[5];
#pragma unroll
    for (int c = 0; c < 5; ++c) {
        const float v = p[c * 32 + lane];
        const bool m  = (v >= LO) && (v <= HI);
        wmask[c] = wave_ballot32(m);          // wave-uniform SGPR result
    }

    // Pack into 3 x u64 (bits 0..159), mask out bit 159 (idx != W-1 in reference).
    unsigned long long x0 = (unsigned long long)wmask[0] |
                            ((unsigned long long)wmask[1] << 32);
    unsigned long long x1 = (unsigned long long)wmask[2] |
                            ((unsigned long long)wmask[3] << 32);
    unsigned long long x2 = (unsigned long long)wmask[4] & 0x7FFFFFFFull; // clear bit 159

    // r = AND of x >> k for k=0..11 over the 192-bit value.
    // Lowest set bit of r == start of first run of >=12 consecutive set bits.
    unsigned long long r0 = x0, r1 = x1, r2 = x2;
#pragma unroll
    for (int k = 1; k <= 11; ++k) {
        const unsigned long long s0 = (x0 >> k) | (x1 << (64 - k));
        const unsigned long long s1 = (x1 >> k) | (x2 << (64 - k));
        const unsigned long long s2 = (x2 >> k);
        r0 &= s0;
        r1 &= s1;
        r2 &= s2;
    }

    int res;
    if (r0)      res = __builtin_ctzll(r0);
    else if (r1) res = 64 + __builtin_ctzll(r1);
    else if (r2) res = 128 + __builtin_ctzll(r2);
    else         res = -1;

    if (lane == 0) out[row] = res;
}

extern "C" void kernel_launch(void* const* d_in, const int* in_sizes, int n_in,
                              void* d_out, int out_size, void* d_ws, size_t ws_size,
                              hipStream_t stream) {
    (void)n_in; (void)out_size; (void)d_ws; (void)ws_size;

    const float* obs = (const float*)d_in[0];
    int* out = (int*)d_out;                       // reference output dtype: int32

    const int n_rows = in_sizes[0] / ROW_STRIDE;  // 32*16 = 512 for the given shape
    const int blocks = (n_rows + 7) / 8;          // 8 waves (rows) per 256-thread block

    BreakoutPlatformPosition_kernel<<<blocks, 256, 0, stream>>>(obs, out, n_rows);
}